// ResidualEMAVectorQuantizer_10230612099577
// MI455X (gfx1250) — compile-verified
//
#include <hip/hip_runtime.h>
#include <hip/hip_bf16.h>
#include <math.h>

typedef __attribute__((ext_vector_type(2))) float v2f;
typedef __attribute__((ext_vector_type(8))) float v8f;

// Problem constants (from reference)
#define NB      16
#define CD      256          // channels == codeword dim
#define HW      1024         // 32*32
#define NROWS   16384        // NB*HW
#define KCB     1024         // codebook size
#define NQ      4

// d_out layout (floats)
#define ZQ_OFF    ((size_t)0)
#define LOSS_POS  ((size_t)4194304)
#define PPL_POS   ((size_t)4194305)
#define ENC_OFF   ((size_t)4194306)
#define IDX_OFF   ((size_t)71303170)

// workspace layout (floats)
#define GRAM_OFF   ((size_t)0)                 // 1024*1024
#define WSQ_OFF    ((size_t)1048576)           // 1024
#define CNT_OFF    ((size_t)(1048576 + 1024))  // 1024
#define LACC_OFF   ((size_t)(1048576 + 2048))  // 1

static __device__ __forceinline__ v8f wmma_f32x4(v2f a, v2f b, v8f c) {
    return __builtin_amdgcn_wmma_f32_16x16x4_f32(false, a, false, b, (short)0, c,
                                                 false, false);
}

// ---------------------------------------------------------------------------
// Gram = W * W^T  [1024 x 1024], fp32 WMMA 16x16x4. 4096 tiles, 8 waves/block.
// ---------------------------------------------------------------------------
__global__ void gram_kernel(const float* __restrict__ Wt, float* __restrict__ gram) {
    const int wave = threadIdx.x >> 5;
    const int lane = threadIdx.x & 31;
    const int lm = lane & 15, half = lane >> 4;
    const int tile = blockIdx.x * 8 + wave;          // [0, 4096)
    const int m0 = (tile >> 6) * 16;
    const int n0 = (tile & 63) * 16;

    const float* arow = Wt + (size_t)(m0 + lm) * CD + 2 * half;
    const float* brow = Wt + (size_t)(n0 + lm) * CD + 2 * half;
    v8f acc = {};
    for (int c = 0; c < CD; c += 4) {
        v2f a = *(const v2f*)(arow + c);
        v2f b = *(const v2f*)(brow + c);
        acc = wmma_f32x4(a, b, acc);
    }
#pragma unroll
    for (int i = 0; i < 8; ++i)
        gram[(size_t)(m0 + i + 8 * half) * KCB + n0 + lm] = acc[i];
}

// ---------------------------------------------------------------------------
// w_sq[k] = sum_c W[k,c]^2.  One wave per k.
// ---------------------------------------------------------------------------
__global__ void wsq_kernel(const float* __restrict__ Wt, float* __restrict__ wsq) {
    const int wave = threadIdx.x >> 5;
    const int lane = threadIdx.x & 31;
    const int k = blockIdx.x * 8 + wave;             // [0, 1024)
    float s = 0.f;
#pragma unroll
    for (int j = 0; j < 8; ++j) {
        float v = Wt[(size_t)k * CD + lane + j * 32];
        s += v * v;
    }
#pragma unroll
    for (int off = 16; off; off >>= 1) s += __shfl_xor(s, off, 32);
    if (lane == 0) wsq[k] = s;
}

// ---------------------------------------------------------------------------
// Fused: zW GEMM (WMMA, scores in registers) + 4 argmin passes via Gram trick
// + all outputs. One block = 16 rows x all 1024 codewords; 8 waves x 8 tiles.
// ---------------------------------------------------------------------------
__global__ void quant_kernel(const float* __restrict__ z,
                             const float* __restrict__ Wt,
                             const float* __restrict__ gram,
                             const float* __restrict__ wsq,
                             float* __restrict__ counts,
                             float* __restrict__ lossAcc,
                             float* __restrict__ out) {
    __shared__ v2f   ldsA[128 * 16];   // [c/2][m] -> {z[2c2], z[2c2+1]} (16 KB)
    __shared__ float ldsRed[256];
    __shared__ float ldsMin[8][16];
    __shared__ int   ldsMinIdx[8][16];
    __shared__ int   ldsSel[NQ][16];
    __shared__ float ldsRsq[16];
    __shared__ float ldsLoss[16];

    const int t = threadIdx.x;
    const int m = t & 15, cg = t >> 4;
    const int n0 = blockIdx.x * 16;
    const int bb = n0 >> 10;           // batch index (tile never crosses batch)
    const int hw0 = n0 & 1023;

    // ---- stage z tile to LDS (coalesced over m), accumulate ||z||^2 ----
    const float* zb = z + (size_t)bb * (CD * HW) + hw0 + m;
    float sq = 0.f;
#pragma unroll
    for (int j = 0; j < 16; ++j) {
        int c = cg + j * 16;
        float v = zb[(size_t)c * HW];
        ((float*)&ldsA[(c >> 1) * 16 + m])[c & 1] = v;
        sq += v * v;
    }
    ldsRed[t] = sq;
    __syncthreads();
    if (t < 16) {
        float s = 0.f;
        for (int j = 0; j < 16; ++j) s += ldsRed[j * 16 + t];
        ldsRsq[t] = s;
        ldsLoss[t] = 0.f;
    }
    __syncthreads();

    // ---- GEMM: acc[nt][i] = r.w for row (i + 8*half), col colBase+nt*16+lm ----
    const int wave = t >> 5, lane = t & 31;
    const int lm = lane & 15, half = lane >> 4;
    const int colBase = wave * 128;

    v8f acc[8] = {};
    for (int c = 0; c < CD; c += 4) {
        v2f a = ldsA[((c >> 1) + half) * 16 + lm];
#pragma unroll
        for (int nt = 0; nt < 8; ++nt) {
            int ncol = colBase + nt * 16 + lm;
            v2f bfr = *(const v2f*)(Wt + (size_t)ncol * CD + c + 2 * half);
            acc[nt] = wmma_f32x4(a, bfr, acc[nt]);
        }
    }

    float wq[8];
#pragma unroll
    for (int nt = 0; nt < 8; ++nt) wq[nt] = wsq[colBase + nt * 16 + lm];

    // ---- 4 residual-quantizer passes ----
    for (int d = 0; d < NQ; ++d) {
        float bv[8];
        int bi[8];
#pragma unroll
        for (int i = 0; i < 8; ++i) { bv[i] = 3.4e38f; bi[i] = 0; }
#pragma unroll
        for (int nt = 0; nt < 8; ++nt) {
            int ncol = colBase + nt * 16 + lm;
#pragma unroll
            for (int i = 0; i < 8; ++i) {
                float s = wq[nt] - 2.0f * acc[nt][i];
                if (s < bv[i] || (s == bv[i] && ncol < bi[i])) { bv[i] = s; bi[i] = ncol; }
            }
        }
        // butterfly across the 16 lanes that share each row
#pragma unroll
        for (int off = 1; off < 16; off <<= 1) {
#pragma unroll
            for (int i = 0; i < 8; ++i) {
                float ov = __shfl_xor(bv[i], off, 32);
                int   oi = __shfl_xor(bi[i], off, 32);
                if (ov < bv[i] || (ov == bv[i] && oi < bi[i])) { bv[i] = ov; bi[i] = oi; }
            }
        }
        if (lm == 0) {
#pragma unroll
            for (int i = 0; i < 8; ++i) {
                ldsMin[wave][half * 8 + i] = bv[i];
                ldsMinIdx[wave][half * 8 + i] = bi[i];
            }
        }
        __syncthreads();
        if (t < 16) {
            float mv = ldsMin[0][t];
            int mi = ldsMinIdx[0][t];
            for (int w2 = 1; w2 < 8; ++w2) {
                float v2 = ldsMin[w2][t];
                int i2 = ldsMinIdx[w2][t];
                if (v2 < mv || (v2 == mv && i2 < mi)) { mv = v2; mi = i2; }
            }
            ldsSel[d][t] = mi;
            // ||r_{d+1}||^2 = ||r_d||^2 + minScore  (minScore = ||w||^2 - 2 r.w)
            float r = ldsRsq[t] + mv;
            ldsRsq[t] = r;
            ldsLoss[t] += r;
        }
        __syncthreads();
        if (d < NQ - 1) {
            // r_new.w = r.w - Gram[sel, :]
#pragma unroll
            for (int i = 0; i < 8; ++i) {
                int row = half * 8 + i;
                const float* grow = gram + (size_t)ldsSel[d][row] * KCB + colBase + lm;
#pragma unroll
                for (int nt = 0; nt < 8; ++nt) acc[nt][i] -= grow[nt * 16];
            }
        }
        __syncthreads();
    }

    // ---- outputs ----
    if (t < 64) {
        int d = t >> 4, mm = t & 15;
        int idx = ldsSel[d][mm];
        int n = n0 + mm;
        out[IDX_OFF + (size_t)(bb * NQ + d) * HW + hw0 + mm] = (float)idx;
        atomicAdd(&counts[idx], 1.0f);
        out[ENC_OFF + ((size_t)d * NROWS + n) * KCB + idx] = 1.0f;
    }
    if (t == 0) {
        float s = 0.f;
        for (int j = 0; j < 16; ++j) s += ldsLoss[j];
        atomicAdd(lossAcc, s);
    }
    // z_q = sum of the 4 selected codewords (straight-through value == quant_sum)
#pragma unroll
    for (int j = 0; j < 16; ++j) {
        int c = cg + j * 16;
        float v = 0.f;
#pragma unroll
        for (int d = 0; d < NQ; ++d) v += Wt[(size_t)ldsSel[d][m] * CD + c];
        out[ZQ_OFF + (size_t)bb * (CD * HW) + (size_t)c * HW + hw0 + m] = v;
    }
}

// ---------------------------------------------------------------------------
// loss + perplexity finalization. 1 block x 1024 threads.
// ---------------------------------------------------------------------------
__global__ void finalize_kernel(const float* __restrict__ counts,
                                const float* __restrict__ lossAcc,
                                float* __restrict__ out) {
    __shared__ float red[1024];
    const int t = threadIdx.x;
    float p = counts[t] * (1.0f / (float)(NROWS * NQ));
    red[t] = p * logf(p + 1e-10f);
    __syncthreads();
    for (int s = 512; s > 0; s >>= 1) {
        if (t < s) red[t] += red[t + s];
        __syncthreads();
    }
    if (t == 0) {
        out[LOSS_POS] = lossAcc[0] * (0.25f / ((float)NROWS * (float)CD));
        out[PPL_POS] = expf(-red[0]);
    }
}

extern "C" void kernel_launch(void* const* d_in, const int* in_sizes, int n_in,
                              void* d_out, int out_size, void* d_ws, size_t ws_size,
                              hipStream_t stream) {
    const float* z  = (const float*)d_in[0];   // [16,256,32,32]
    const float* Wt = (const float*)d_in[1];   // [1024,256]
    float* out = (float*)d_out;
    float* ws = (float*)d_ws;

    float* gram    = ws + GRAM_OFF;
    float* wsq     = ws + WSQ_OFF;
    float* counts  = ws + CNT_OFF;
    float* lossAcc = ws + LACC_OFF;

    // zero accumulators + the 256 MB one-hot encodings region (capturable)
    hipMemsetAsync(counts, 0, (1024 + 1) * sizeof(float), stream);
    hipMemsetAsync(out + ENC_OFF, 0, (size_t)NROWS * NQ * KCB * sizeof(float), stream);

    gram_kernel<<<512, 256, 0, stream>>>(Wt, gram);
    wsq_kernel<<<128, 256, 0, stream>>>(Wt, wsq);
    quant_kernel<<<NROWS / 16, 256, 0, stream>>>(z, Wt, gram, wsq, counts, lossAcc, out);
    finalize_kernel<<<1, 1024, 0, stream>>>(counts, lossAcc, out);
}